// GCN_120259084570
// MI455X (gfx1250) — compile-verified
//
#include <hip/hip_runtime.h>
#include <hip/hip_bf16.h>

typedef __attribute__((ext_vector_type(2))) float v2f;
typedef __attribute__((ext_vector_type(8))) float v8f;

#define GCN_FEATS 128

// ---------------------------------------------------------------------------
// Degree accumulation: deg_out[src[e]] += 1, deg_in[dst[e]] += 1
// ---------------------------------------------------------------------------
__global__ void gcn_degrees(const int* __restrict__ src, const int* __restrict__ dst,
                            float* __restrict__ degOut, float* __restrict__ degIn,
                            int nEdges) {
    int e = blockIdx.x * blockDim.x + threadIdx.x;
    if (e < nEdges) {
        atomicAdd(&degOut[src[e]], 1.0f);
        atomicAdd(&degIn[dst[e]], 1.0f);
    }
}

// In-place deg -> rsqrt(max(deg,1))
__global__ void gcn_rsqrt(float* __restrict__ d, int n) {
    int i = blockIdx.x * blockDim.x + threadIdx.x;
    if (i < n) d[i] = rsqrtf(fmaxf(d[i], 1.0f));
}

// ---------------------------------------------------------------------------
// Edge aggregation: one wave32 per edge, lane handles 4 consecutive feats.
// Out[dst[e]][:] += (SCALE ? scale[src[e]] : 1) * H[src[e]][:]
// Feature rows (25.6 MB) are L2-resident; atomics land in L2 atomic units.
// ---------------------------------------------------------------------------
template <bool SCALE>
__global__ void gcn_aggregate(const float* __restrict__ H,
                              const float* __restrict__ scale,
                              const int* __restrict__ src, const int* __restrict__ dst,
                              float* __restrict__ Out, int nEdges) {
    int wave = (int)((blockIdx.x * blockDim.x + threadIdx.x) >> 5);
    int lane = threadIdx.x & 31;
    if (wave >= nEdges) return;
    int s = src[wave];
    int d = dst[wave];
    const float4* hp = (const float4*)(H + (size_t)s * GCN_FEATS);
    float4 v = hp[lane];
    if (SCALE) {
        float sc = scale[s];
        v.x *= sc; v.y *= sc; v.z *= sc; v.w *= sc;
    }
    float* op = Out + (size_t)d * GCN_FEATS + lane * 4;
    atomicAdd(op + 0, v.x);
    atomicAdd(op + 1, v.y);
    atomicAdd(op + 2, v.z);
    atomicAdd(op + 3, v.w);
}

// ---------------------------------------------------------------------------
// Fused GEMM: Out = f(((A * rowScale) @ W) + bias), f = relu*postScale or id.
// One wave per 16x16 C tile via V_WMMA_F32_16X16X4_F32 (full fp32 — GEMM is
// ~2.5 GFLOP total, nowhere near the bottleneck, so keep reference precision).
// A tile (16 x 128, pre-scaled by norm_dst) staged in LDS, shared by all
// waves in the block; each wave owns one 16-col slice of N.
// ---------------------------------------------------------------------------
template <int NCOLS, bool RELU_SCALE>
__global__ void gcn_gemm_wmma(const float* __restrict__ A,
                              const float* __restrict__ rowScale,   // norm_dst
                              const float* __restrict__ W,          // [128, NCOLS]
                              const float* __restrict__ bias,       // [NCOLS]
                              const float* __restrict__ postScale,  // norm_src (if RELU_SCALE)
                              float* __restrict__ Out,              // [M, NCOLS]
                              int M) {
    constexpr int K = GCN_FEATS;
    constexpr int NWAVES = NCOLS / 16;
    __shared__ float As[16 * K];  // 8 KB

    const int row0 = blockIdx.x * 16;
    const int tid = threadIdx.x;

    // Cooperative, coalesced load of the A tile, fused with norm_dst scaling.
    for (int i = tid; i < 16 * K; i += NWAVES * 32) {
        int r = i / K, c = i % K;
        int gr = row0 + r;
        float v = 0.0f;
        if (gr < M) v = A[(size_t)gr * K + c] * rowScale[gr];
        As[i] = v;
    }
    __syncthreads();

    const int wave = tid >> 5;
    const int lane = tid & 31;
    const int half = lane >> 4;   // 0: K-pair {0,1} / C rows 0-7; 1: K-pair {2,3} / C rows 8-15
    const int l16  = lane & 15;
    const int n0   = wave * 16;

    v8f c = {0.f, 0.f, 0.f, 0.f, 0.f, 0.f, 0.f, 0.f};
    const float* arow = &As[l16 * K];

#pragma unroll
    for (int k = 0; k < K; k += 4) {
        v2f a, b;
        // A 16x4 fragment: lane row = l16, K offsets = k + 2*half + {0,1}
        a.x = arow[k + 2 * half + 0];
        a.y = arow[k + 2 * half + 1];
        // B 4x16 fragment: (K = k + 2*half + {0,1}, N = n0 + l16), W row-major [K,N]
        const float* wp = &W[(size_t)(k + 2 * half) * NCOLS + n0 + l16];
        b.x = wp[0];
        b.y = wp[NCOLS];
        c = __builtin_amdgcn_wmma_f32_16x16x4_f32(false, a, false, b, (short)0, c,
                                                  false, false);
    }

    // C layout: VGPR r -> row = row0 + r + 8*half, col = n0 + l16
    const int col = n0 + l16;
    const float bv = bias[col];
#pragma unroll
    for (int r = 0; r < 8; ++r) {
        int gr = row0 + r + 8 * half;
        if (gr < M) {
            float v = c[r] + bv;
            if constexpr (RELU_SCALE) {
                v = fmaxf(v, 0.0f) * postScale[gr];  // fuse relu + norm_src for layer 2
            }
            Out[(size_t)gr * NCOLS + col] = v;
        }
    }
}

// ---------------------------------------------------------------------------
extern "C" void kernel_launch(void* const* d_in, const int* in_sizes, int n_in,
                              void* d_out, int out_size, void* d_ws, size_t ws_size,
                              hipStream_t stream) {
    const float* x  = (const float*)d_in[0];
    const float* W1 = (const float*)d_in[1];
    const float* b1 = (const float*)d_in[2];
    const float* W2 = (const float*)d_in[3];
    const float* b2 = (const float*)d_in[4];
    const int* src  = (const int*)d_in[5];
    const int* dst  = (const int*)d_in[6];

    const int nNodes = in_sizes[0] / GCN_FEATS;
    const int nEdges = in_sizes[5];

    char* ws = (char*)d_ws;
    float* deg = (float*)ws;              // [2*nNodes]: deg_out | deg_in -> norms in place
    float* normSrc = deg;
    float* normDst = deg + nNodes;
    size_t degBytes = (size_t)2 * nNodes * sizeof(float);
    size_t off = (degBytes + 255) & ~(size_t)255;
    size_t featBytes = (size_t)nNodes * GCN_FEATS * sizeof(float);
    float* bufA = (float*)(ws + off);                                   // aggregation accum
    float* bufB = (float*)(ws + off + ((featBytes + 255) & ~(size_t)255));  // h1 * norm_src

    // Zero degree accumulators + first aggregation buffer (graph-capture safe).
    hipMemsetAsync(deg, 0, degBytes, stream);
    hipMemsetAsync(bufA, 0, featBytes, stream);

    gcn_degrees<<<(nEdges + 255) / 256, 256, 0, stream>>>(src, dst, normSrc, normDst, nEdges);
    gcn_rsqrt<<<(2 * nNodes + 255) / 256, 256, 0, stream>>>(deg, 2 * nNodes);

    // Layer 1: m = scatter(x * norm_src); h1s = relu(m*norm_dst @ W1 + b1) * norm_src
    gcn_aggregate<true><<<(nEdges + 7) / 8, 256, 0, stream>>>(x, normSrc, src, dst, bufA, nEdges);
    gcn_gemm_wmma<128, true><<<(nNodes + 15) / 16, 256, 0, stream>>>(
        bufA, normDst, W1, b1, normSrc, bufB, nNodes);

    // Layer 2: m2 = scatter(h1s); out = m2*norm_dst @ W2 + b2
    hipMemsetAsync(bufA, 0, featBytes, stream);
    gcn_aggregate<false><<<(nEdges + 7) / 8, 256, 0, stream>>>(bufB, nullptr, src, dst, bufA, nEdges);
    gcn_gemm_wmma<64, false><<<(nNodes + 15) / 16, 128, 0, stream>>>(
        bufA, normDst, W2, b2, nullptr, (float*)d_out, nNodes);
}